// Pooled_Cross_Attention_28587302322251
// MI455X (gfx1250) — compile-verified
//
#include <hip/hip_runtime.h>

typedef float v2f __attribute__((ext_vector_type(2)));
typedef float v4f __attribute__((ext_vector_type(4)));
typedef float v8f __attribute__((ext_vector_type(8)));

#define DIM      4096   // model dim == n_heads*head_dim
#define UP_DIM   1024   // n_kv_heads*head_dim
#define B_ROWS   8      // batch
#define TOTAL_T  16384  // sum(seqlen)

// ---------------------------------------------------------------------------
// One wave accumulates a 16x16 f32 tile of  A(16 x K) * W^T  over [k_begin,k_end)
// using V_WMMA_F32_16X16X4_F32 (full f32 precision, matches reference dtype).
//   A: row-major [16 x lda]; rows >= a_rows contribute zero (mask-mul, no branch
//      so EXEC stays all-ones as WMMA requires)
//   W: row-major [N x ldw]  (nn.Linear weight); tile column n <-> W row n_base+n
// ISA 16x16x4 f32 A layout: lanes 0-15 hold {K+0,K+1}, lanes 16-31 hold {K+2,K+3}
// -> contiguous float2 per lane per K-step. B assumed symmetric (N across lanes).
// ---------------------------------------------------------------------------
__device__ inline v8f wmma_tile_f32(const float* __restrict__ A, int lda, int a_rows,
                                    const float* __restrict__ W, int ldw,
                                    int n_base, int k_begin, int k_end)
{
    const int lane  = threadIdx.x & 31;
    const int half  = lane >> 4;            // 0: K+0..1   1: K+2..3
    const int m     = lane & 15;            // A row this lane feeds
    const float amask = (m < a_rows) ? 1.0f : 0.0f;
    const int   am    = (m < a_rows) ? m : 0;     // safe address for padded rows
    const float* __restrict__ arow = A + (size_t)am * lda;
    const float* __restrict__ wrow = W + (size_t)(n_base + m) * ldw;

    v8f acc = {};
    #pragma unroll 8
    for (int k = k_begin; k < k_end; k += 4) {
        const int ko = k + 2 * half;
        v2f a = *(const v2f*)(arow + ko);
        a.x *= amask;
        a.y *= amask;
        v2f b = *(const v2f*)(wrow + ko);
        // (neg_a, A, neg_b, B, c_mod, C, reuse_a, reuse_b)
        acc = __builtin_amdgcn_wmma_f32_16x16x4_f32(
            false, a, false, b, (short)0, acc, false, false);
    }
    return acc;
}

// ---------------------------------------------------------------------------
// Kernel 1: x = embedding @ W_up.T  ([8,1024]); store directly as the
// kv-head repeat-interleaved val [16 x 4096] (rows 8..15 zeroed for padding).
// Grid: 64 blocks (one 16-col tile each) x 128 threads (4 waves split K=4096).
// ---------------------------------------------------------------------------
__global__ void __launch_bounds__(128)
upproj_repeat_kernel(const float* __restrict__ emb, const float* __restrict__ Wup,
                     float* __restrict__ val /* 16 x 4096 */)
{
    const int wave = threadIdx.x >> 5;
    const int lane = threadIdx.x & 31;
    const int n_base = blockIdx.x * 16;               // column tile in [0,1024)
    const int kc = DIM / 4;                           // 1024 K per wave

    v8f acc = wmma_tile_f32(emb, DIM, B_ROWS, Wup, DIM, n_base,
                            wave * kc, (wave + 1) * kc);

    __shared__ float red[4][32][8];
    #pragma unroll
    for (int r = 0; r < 8; ++r) red[wave][lane][r] = acc[r];
    __syncthreads();

    if (wave == 0) {
        #pragma unroll
        for (int r = 0; r < 8; ++r)
            acc[r] = red[0][lane][r] + red[1][lane][r] + red[2][lane][r] + red[3][lane][r];
        // C layout: VGPR r = rows {r (lanes 0-15), r+8 (lanes 16-31)}, N = lane&15
        const int nc  = n_base + (lane & 15);         // x column 0..1023
        const int kv  = nc >> 7;                      // kv head
        const int c   = nc & 127;                     // within head
        const int mr0 = (lane < 16) ? 0 : 8;
        #pragma unroll
        for (int r = 0; r < 8; ++r) {
            const float v = (lane < 16) ? acc[r] : 0.0f;  // rows 8-15: zero pad
            const size_t rowoff = (size_t)(mr0 + r) * DIM;
            #pragma unroll
            for (int rep = 0; rep < 4; ++rep)             // repeat_interleave x4
                val[rowoff + (size_t)((kv * 4 + rep) * 128 + c)] = v;
        }
    }
}

// ---------------------------------------------------------------------------
// Kernel 2: y = val @ W_down.T  ([8,4096]); the repeat over tokens factors out
// of the matmul, so only 8 unique rows are projected (268 MFLOP vs 550 GFLOP).
// Grid: 256 blocks (16-col tiles of N=4096) x 128 threads (4 waves split K).
// ---------------------------------------------------------------------------
__global__ void __launch_bounds__(128)
downproj_kernel(const float* __restrict__ val, const float* __restrict__ Wdown,
                float* __restrict__ y /* 8 x 4096 */)
{
    const int wave = threadIdx.x >> 5;
    const int lane = threadIdx.x & 31;
    const int n_base = blockIdx.x * 16;               // column tile in [0,4096)
    const int kc = DIM / 4;

    v8f acc = wmma_tile_f32(val, DIM, 16, Wdown, DIM, n_base,
                            wave * kc, (wave + 1) * kc);

    __shared__ float red[4][32][8];
    #pragma unroll
    for (int r = 0; r < 8; ++r) red[wave][lane][r] = acc[r];
    __syncthreads();

    if (wave == 0 && lane < 16) {                     // rows 0-7 live in lanes 0-15
        const int nc = n_base + lane;
        #pragma unroll
        for (int r = 0; r < 8; ++r) {
            const float v = red[0][lane][r] + red[1][lane][r]
                          + red[2][lane][r] + red[3][lane][r];
            y[(size_t)r * DIM + nc] = v;
        }
    }
}

// ---------------------------------------------------------------------------
// Kernel 3: ragged row broadcast — out[t,:] = y[row(t),:].  256 MB streaming
// write == the whole-problem bound (~11 us at 23.3 TB/s).  One block per token
// row; y (128 KB) stays L2-hot; output is write-once -> nontemporal b128 stores.
// ---------------------------------------------------------------------------
__global__ void __launch_bounds__(256)
broadcast_kernel(const float* __restrict__ y, const int* __restrict__ seqlen,
                 float* __restrict__ out)
{
    const int t = blockIdx.x;
    int cum = 0, row = 0;
    #pragma unroll
    for (int i = 0; i < 8; ++i) {                     // branchless prefix search
        cum += seqlen[i];
        row = (t >= cum) ? (i + 1) : row;
    }

    const v4f* __restrict__ src = (const v4f*)(y + (size_t)row * DIM);
    v4f* __restrict__ dst = (v4f*)(out + (size_t)t * DIM);
    const int tid = threadIdx.x;
    #pragma unroll
    for (int j = 0; j < 4; ++j) {                     // 4096 floats = 4 x 256 v4f
        const int i4 = tid + j * 256;
        v4f v = src[i4];
        __builtin_nontemporal_store(v, &dst[i4]);
    }
}

// ---------------------------------------------------------------------------
extern "C" void kernel_launch(void* const* d_in, const int* in_sizes, int n_in,
                              void* d_out, int out_size, void* d_ws, size_t ws_size,
                              hipStream_t stream)
{
    const float* emb    = (const float*)d_in[0];   // [8, 4096]
    const float* Wup    = (const float*)d_in[1];   // [1024, 4096]
    const float* Wdown  = (const float*)d_in[2];   // [4096, 4096]
    const int*   seqlen = (const int*)d_in[3];     // [8]
    float* out = (float*)d_out;                    // [16384, 4096]

    float* val = (float*)d_ws;                     // 16 x 4096 f32 (256 KB)
    float* y   = val + 16 * DIM;                   //  8 x 4096 f32 (128 KB)

    upproj_repeat_kernel<<<UP_DIM / 16, 128, 0, stream>>>(emb, Wup, val);
    downproj_kernel<<<DIM / 16, 128, 0, stream>>>(val, Wdown, y);
    broadcast_kernel<<<TOTAL_T, 256, 0, stream>>>(y, seqlen, out);
}